// HSCRF_27556510171565
// MI455X (gfx1250) — compile-verified
//
#include <hip/hip_runtime.h>
#include <math.h>

// Problem constants (from reference setup_inputs: B=16, L=128, T=11, K=7)
#define B_       16
#define L_       128
#define T_       11
#define K_       7
#define TT_      121          // T*T
#define START_ID 10           // T-1
#define STOP_ID  9            // T-2
#define NEGV     (-10000.0f)
#define MASKNEG  (-1.0e9f)
// Async prefetch pipeline depth (power of 2). 8 slabs * 7 async instrs/wave
// = 56 outstanding, within the 6-bit ASYNCcnt capacity (63).
#define DEPTH    8

// ---- CDNA5 async global->LDS path (gated; fallback = synchronous loads) ----
#if defined(__gfx1250__) && __has_builtin(__builtin_amdgcn_global_load_async_to_lds_b32)
#define HAVE_ASYNC 1
#else
#define HAVE_ASYNC 0
#endif

#if HAVE_ASYNC
#if __has_builtin(__builtin_amdgcn_s_wait_asynccnt)
#define WAIT_ASYNC(N) __builtin_amdgcn_s_wait_asynccnt((unsigned short)(N))
#else
#define WAIT_ASYNC(N) asm volatile("s_wait_asynccnt %0" ::"n"(N) : "memory")
#endif
// Builtin prototype (per hipcc diagnostic): param 0 is a non-const
// address-space(1) int* ("__device__ int *"); param 1 is the LDS (as3) dest.
typedef __attribute__((address_space(1))) int GAS_i32;
typedef __attribute__((address_space(3))) int LAS_i32;
#endif

// One workgroup per batch element. 128 threads = 4 wave32s.
// LDS holds: full alpha table, a DEPTH-deep ring of score slabs (7x121 floats
// per DP step), and 121 logsumexp partials for the cross-y_prev reduction.
// Total LDS ~34 KB (alpha 5.7KB + ring 27.1KB + partials 1KB) << 320 KB/WGP.
__global__ __launch_bounds__(128) void semicrf_fwd_kernel(
    const float* __restrict__ scores,  // (B, L, L, T, T)
    const int* __restrict__ mask,      // (B,)
    float* __restrict__ partial)       // (B,) per-batch alpha[mask][stop]
{
  __shared__ float s_alpha[(L_ + 1) * T_];       // 1419 floats
  __shared__ float s_slab[DEPTH][K_ * TT_];      // DEPTH * 847 floats
  __shared__ float s_m[TT_];
  __shared__ float s_s[TT_];

  const int b = blockIdx.x;
  const int t = threadIdx.x;
  const float* __restrict__ sb = scores + (size_t)b * L_ * L_ * TT_;

  const int yp = t / T_;        // meaningful for t < 121
  const int yy = t - yp * T_;
  (void)yy;

  // alpha row 0: NEG everywhere except start label
  if (t < T_) s_alpha[t] = (t == START_ID) ? 0.0f : NEGV;

#if HAVE_ASYNC
  // Issue one DP step's 7x121-float slab as 7 EXEC-masked async b32 loads per
  // wave (uniform count per wave -> ASYNCcnt bookkeeping is wave-uniform).
  auto issue_slab = [&](int i, int slot) {
    const int e = i - 1;
#pragma unroll
    for (int k = 0; k < K_; ++k) {
      const int j  = i - K_ + k;
      const int jc = (j < 0) ? 0 : j;          // clamp: garbage rows are masked in compute
      const float* src = sb + ((size_t)jc * L_ + e) * TT_ + t;
      if (t < TT_) {
        __builtin_amdgcn_global_load_async_to_lds_b32(
            (GAS_i32*)src,
            (LAS_i32*)&s_slab[slot][k * TT_ + t],
            /*offset=*/0, /*cpol=*/0);
      }
    }
  };
  // Prologue: fill the pipeline with slabs for steps 1..DEPTH.
  for (int i = 1; i <= DEPTH; ++i) issue_slab(i, (i - 1) & (DEPTH - 1));
#endif

  for (int i = 1; i <= L_; ++i) {
    const int slot = (i - 1) & (DEPTH - 1);

#if HAVE_ASYNC
    // Slab for step i is the oldest in flight; in-order completion means
    // outstanding <= (DEPTH-1)*K_ guarantees it has landed in LDS.
    WAIT_ASYNC((DEPTH - 1) * K_);
    __syncthreads();   // slab visible to all waves; alpha[i-1] visible too
    if (i + DEPTH <= L_) issue_slab(i + DEPTH, (i + DEPTH - 1) & (DEPTH - 1));
#else
    {
      const int e = i - 1;
      if (t < TT_) {
#pragma unroll
        for (int k = 0; k < K_; ++k) {
          const int j  = i - K_ + k;
          const int jc = (j < 0) ? 0 : j;
          s_slab[slot][k * TT_ + t] = sb[((size_t)jc * L_ + e) * TT_ + t];
        }
      }
      __syncthreads();
    }
#endif

    // Stage 1: per (y_prev, y) pair, stable logsumexp over the K=7 span starts.
    if (t < TT_) {
      float v[K_];
#pragma unroll
      for (int k = 0; k < K_; ++k) {
        const int j  = i - K_ + k;
        const int jc = (j < 0) ? 0 : j;
        const float sv = s_slab[slot][k * TT_ + t];
        v[k] = (j >= 0) ? (sv + s_alpha[jc * T_ + yp]) : MASKNEG;
      }
      float m = v[0];
#pragma unroll
      for (int k = 1; k < K_; ++k) m = fmaxf(m, v[k]);
      float s = 0.0f;
#pragma unroll
      for (int k = 0; k < K_; ++k) s += __expf(v[k] - m);  // MASKNEG terms -> 0
      s_m[t] = m;
      s_s[t] = s;
    }
    __syncthreads();

    // Stage 2: combine 11 y_prev partials per output label y -> alpha[i][y].
    if (t < T_) {
      float m2 = s_m[t];
#pragma unroll
      for (int p = 1; p < T_; ++p) m2 = fmaxf(m2, s_m[p * T_ + t]);
      float acc = 0.0f;
#pragma unroll
      for (int p = 0; p < T_; ++p)
        acc += s_s[p * T_ + t] * __expf(s_m[p * T_ + t] - m2);
      s_alpha[i * T_ + t] = m2 + __logf(acc);
    }
    // No trailing barrier needed: the barrier at the top of the next
    // iteration orders stage-2 writes against stage-1 reads / s_m reuse.
  }

  __syncthreads();
  if (t == 0) {
    int mlen = mask[b];
    mlen = (mlen < 0) ? 0 : (mlen > L_ ? L_ : mlen);
    partial[b] = s_alpha[mlen * T_ + STOP_ID];
  }
}

// Deterministic final reduction of 16 per-batch partials (avoids atomics into
// the poisoned d_out; same-stream ordering makes this graph-safe).
__global__ void semicrf_reduce_kernel(const float* __restrict__ partial,
                                      float* __restrict__ out)
{
  if (blockIdx.x == 0 && threadIdx.x == 0) {
    float s = 0.0f;
#pragma unroll
    for (int b = 0; b < B_; ++b) s += partial[b];
    out[0] = s;
  }
}

extern "C" void kernel_launch(void* const* d_in, const int* in_sizes, int n_in,
                              void* d_out, int out_size, void* d_ws, size_t ws_size,
                              hipStream_t stream) {
  const float* scores = (const float*)d_in[0];  // (16,128,128,11,11) f32
  const int*   mask   = (const int*)d_in[1];    // (16,) i32
  float* partial = (float*)d_ws;                // 16 floats of scratch
  float* out = (float*)d_out;                   // scalar f32

  semicrf_fwd_kernel<<<B_, 128, 0, stream>>>(scores, mask, partial);
  semicrf_reduce_kernel<<<1, 32, 0, stream>>>(partial, out);
}